// Decoder_20083267076335
// MI455X (gfx1250) — compile-verified
//
#include <hip/hip_runtime.h>
#include <math.h>

typedef __attribute__((ext_vector_type(2))) float v2f;
typedef __attribute__((ext_vector_type(8))) float v8f;

#define E_DIM 64
#define H_DIM 128
#define O_DIM 16
#define BATCH 64
#define NNODES 1000
#define FUT 12
#define G4 512   // 4*H

__device__ __forceinline__ float sigm(float x) { return 1.0f / (1.0f + __expf(-x)); }

__global__ __launch_bounds__(256) void decoder_scan(
    const float* __restrict__ enc,   // (B,1,N,E)
    const float* __restrict__ mean_, // (B,FUT,N,O)
    const float* __restrict__ s,     // (1,O)
    const float* __restrict__ h0,    // (2,O)
    const float* __restrict__ c0,    // (2,H)
    const float* __restrict__ Wih0,  // (512,80)
    const float* __restrict__ Whh0,  // (512,16)
    const float* __restrict__ bih0, const float* __restrict__ bhh0,
    const float* __restrict__ Whr0,  // (16,128)
    const float* __restrict__ Wih1,  // (512,16)
    const float* __restrict__ Whh1,  // (512,16)
    const float* __restrict__ bih1, const float* __restrict__ bhh1,
    const float* __restrict__ Whr1,  // (16,128)
    float* __restrict__ out)         // (B,FUT,N,O)
{
  // ---- LDS working set (~205 KB, < 320 KB/WGP) ----
  __shared__ float WcatT0[32][G4];   // rows 0-15: Wih0[:,64:80]^T ; rows 16-31: Whh0^T
  __shared__ float WcatT1[32][G4];   // rows 0-15: Wih1^T ; rows 16-31: Whh1^T
  __shared__ float Whr0T[H_DIM][O_DIM];
  __shared__ float Whr1T[H_DIM][O_DIM];
  __shared__ float GencB[16][G4];    // per-node hoisted enc@Wenc^T + bias0
  __shared__ float bias0s[G4];
  __shared__ float bias1s[G4];
  __shared__ float hfull[16][H_DIM];
  __shared__ float part[8][16][16];
  __shared__ float xin0[16][32];
  __shared__ float xin1[16][32];
  __shared__ float h1Buf[16][16];
  __shared__ float h2Buf[16][16];
  __shared__ float lastBuf[16][16];
  __shared__ float nmBuf[16][16];

  const int tid  = threadIdx.x;
  const int w    = tid >> 5;     // wave id 0..7
  const int lane = tid & 31;
  const int lm   = lane & 15;    // A row / B,C col within tile
  const int lh   = lane >> 4;    // half-wave select
  const int b0   = blockIdx.x * 16;

  // ---- stage weights into LDS (transposed), fold biases ----
  for (int idx = tid; idx < 32 * G4; idx += 256) {
    int k = idx >> 9;            // 0..31
    int col = idx & 511;
    WcatT0[k][col] = (k < 16) ? Wih0[col * 80 + 64 + k] : Whh0[col * 16 + (k - 16)];
    WcatT1[k][col] = (k < 16) ? Wih1[col * 16 + k]      : Whh1[col * 16 + (k - 16)];
  }
  for (int idx = tid; idx < H_DIM * O_DIM; idx += 256) {
    int k = idx >> 4; int col = idx & 15;
    Whr0T[k][col] = Whr0[col * H_DIM + k];
    Whr1T[k][col] = Whr1[col * H_DIM + k];
  }
  for (int col = tid; col < G4; col += 256) {
    bias0s[col] = bih0[col] + bhh0[col];
    bias1s[col] = bih1[col] + bhh1[col];
  }
  {
    int row = tid >> 4, o = tid & 15;
    h1Buf[row][o]   = h0[o];
    h2Buf[row][o]   = h0[O_DIM + o];
    lastBuf[row][o] = s[o];
  }
  // cell states live in WMMA-accumulator layout registers for the whole scan
  const int hcol = 16 * w + lm;
  float c1r[8], c2r[8];
  #pragma unroll
  for (int r = 0; r < 8; ++r) { c1r[r] = c0[hcol]; c2r[r] = c0[H_DIM + hcol]; }
  __syncthreads();

  for (int n = 0; n < NNODES; ++n) {
    // ---- node mean (B,FUT,N,O) averaged over FUT ----
    {
      int row = tid >> 4, o = tid & 15;
      const float* mp = mean_ + (size_t)(b0 + row) * FUT * NNODES * O_DIM
                              + (size_t)n * O_DIM + o;
      float sum = 0.f;
      #pragma unroll
      for (int f = 0; f < FUT; ++f) sum += mp[(size_t)f * NNODES * O_DIM];
      nmBuf[row][o] = sum * (1.0f / FUT);
    }
    // ---- hoist enc contribution: GencB = enc_tile(16x64) @ Wenc^T + bias0 ----
    #pragma unroll
    for (int g = 0; g < 4; ++g) {
      int gc = 16 * (w + 8 * g) + lm;
      v8f acc;
      float bv = bias0s[gc];
      #pragma unroll
      for (int r = 0; r < 8; ++r) acc[r] = bv;
      const float* ap = enc + (size_t)(b0 + lm) * NNODES * E_DIM + (size_t)n * E_DIM;
      #pragma unroll
      for (int k = 0; k < 16; ++k) {
        int kk = 4 * k + 2 * lh;
        v2f a, b;
        a.x = ap[kk];              a.y = ap[kk + 1];
        b.x = Wih0[gc * 80 + kk];  b.y = Wih0[gc * 80 + kk + 1];
        acc = __builtin_amdgcn_wmma_f32_16x16x4_f32(false, a, false, b, (short)0, acc, false, false);
      }
      #pragma unroll
      for (int r = 0; r < 8; ++r) GencB[r + 8 * lh][gc] = acc[r];
    }
    __syncthreads();

    for (int f = 0; f < FUT; ++f) {
      // xin0 = [last + nm | h1_prev]  (16x32)
      {
        int row = tid >> 4, o = tid & 15;
        xin0[row][o]      = lastBuf[row][o] + nmBuf[row][o];
        xin0[row][16 + o] = h1Buf[row][o];
      }
      __syncthreads();

      // ---- layer 0: gates = GencB + xin0 @ [Wloop;Whh0]^T ; cell update ----
      {
        v8f acc[4];
        #pragma unroll
        for (int g = 0; g < 4; ++g) {
          int gc = 16 * (w + 8 * g) + lm;
          #pragma unroll
          for (int r = 0; r < 8; ++r) acc[g][r] = GencB[r + 8 * lh][gc];
        }
        #pragma unroll
        for (int k = 0; k < 8; ++k) {
          int kk = 4 * k + 2 * lh;
          v2f a; a.x = xin0[lm][kk]; a.y = xin0[lm][kk + 1];
          #pragma unroll
          for (int g = 0; g < 4; ++g) {
            int gc = 16 * (w + 8 * g) + lm;
            v2f b; b.x = WcatT0[kk][gc]; b.y = WcatT0[kk + 1][gc];
            acc[g] = __builtin_amdgcn_wmma_f32_16x16x4_f32(false, a, false, b, (short)0, acc[g], false, false);
          }
        }
        #pragma unroll
        for (int r = 0; r < 8; ++r) {
          float iv = sigm(acc[0][r]);
          float fv = sigm(acc[1][r]);
          float gv = tanhf(acc[2][r]);
          float ov = sigm(acc[3][r]);
          float cn = fv * c1r[r] + iv * gv;
          c1r[r] = cn;
          hfull[r + 8 * lh][hcol] = ov * tanhf(cn);
        }
      }
      __syncthreads();

      // ---- h1 = hfull(16x128) @ Whr0^T : K split across 8 waves, LDS reduce ----
      {
        v8f acc = {};
        #pragma unroll
        for (int k = 0; k < 4; ++k) {
          int kk = 16 * w + 4 * k + 2 * lh;
          v2f a; a.x = hfull[lm][kk];  a.y = hfull[lm][kk + 1];
          v2f b; b.x = Whr0T[kk][lm];  b.y = Whr0T[kk + 1][lm];
          acc = __builtin_amdgcn_wmma_f32_16x16x4_f32(false, a, false, b, (short)0, acc, false, false);
        }
        #pragma unroll
        for (int r = 0; r < 8; ++r) part[w][r + 8 * lh][lm] = acc[r];
      }
      __syncthreads();
      {
        int row = tid >> 4, o = tid & 15;
        float v = 0.f;
        #pragma unroll
        for (int p = 0; p < 8; ++p) v += part[p][row][o];
        h1Buf[row][o] = v;
        xin1[row][o] = v;
        xin1[row][16 + o] = h2Buf[row][o];
      }
      __syncthreads();

      // ---- layer 1: gates = bias1 + [h1|h2] @ [Wih1;Whh1]^T ; cell update ----
      {
        v8f acc[4];
        #pragma unroll
        for (int g = 0; g < 4; ++g) {
          int gc = 16 * (w + 8 * g) + lm;
          float bv = bias1s[gc];
          #pragma unroll
          for (int r = 0; r < 8; ++r) acc[g][r] = bv;
        }
        #pragma unroll
        for (int k = 0; k < 8; ++k) {
          int kk = 4 * k + 2 * lh;
          v2f a; a.x = xin1[lm][kk]; a.y = xin1[lm][kk + 1];
          #pragma unroll
          for (int g = 0; g < 4; ++g) {
            int gc = 16 * (w + 8 * g) + lm;
            v2f b; b.x = WcatT1[kk][gc]; b.y = WcatT1[kk + 1][gc];
            acc[g] = __builtin_amdgcn_wmma_f32_16x16x4_f32(false, a, false, b, (short)0, acc[g], false, false);
          }
        }
        #pragma unroll
        for (int r = 0; r < 8; ++r) {
          float iv = sigm(acc[0][r]);
          float fv = sigm(acc[1][r]);
          float gv = tanhf(acc[2][r]);
          float ov = sigm(acc[3][r]);
          float cn = fv * c2r[r] + iv * gv;
          c2r[r] = cn;
          hfull[r + 8 * lh][hcol] = ov * tanhf(cn);
        }
      }
      __syncthreads();

      // ---- h2 = hfull @ Whr1^T : same split-K + reduce; write state + output ----
      {
        v8f acc = {};
        #pragma unroll
        for (int k = 0; k < 4; ++k) {
          int kk = 16 * w + 4 * k + 2 * lh;
          v2f a; a.x = hfull[lm][kk];  a.y = hfull[lm][kk + 1];
          v2f b; b.x = Whr1T[kk][lm];  b.y = Whr1T[kk + 1][lm];
          acc = __builtin_amdgcn_wmma_f32_16x16x4_f32(false, a, false, b, (short)0, acc, false, false);
        }
        #pragma unroll
        for (int r = 0; r < 8; ++r) part[w][r + 8 * lh][lm] = acc[r];
      }
      __syncthreads();
      {
        int row = tid >> 4, o = tid & 15;
        float v = 0.f;
        #pragma unroll
        for (int p = 0; p < 8; ++p) v += part[p][row][o];
        h2Buf[row][o]   = v;
        lastBuf[row][o] = v;
        out[(size_t)(b0 + row) * FUT * NNODES * O_DIM
            + (size_t)f * NNODES * O_DIM + (size_t)n * O_DIM + o] = v + nmBuf[row][o];
      }
      __syncthreads();
    }
  }
}

extern "C" void kernel_launch(void* const* d_in, const int* in_sizes, int n_in,
                              void* d_out, int out_size, void* d_ws, size_t ws_size,
                              hipStream_t stream) {
  (void)in_sizes; (void)n_in; (void)out_size; (void)d_ws; (void)ws_size;
  const float* enc   = (const float*)d_in[0];
  const float* mean_ = (const float*)d_in[1];
  const float* s     = (const float*)d_in[2];
  const float* h0    = (const float*)d_in[3];
  const float* c0    = (const float*)d_in[4];
  const float* Wih0  = (const float*)d_in[5];
  const float* Whh0  = (const float*)d_in[6];
  const float* bih0  = (const float*)d_in[7];
  const float* bhh0  = (const float*)d_in[8];
  const float* Whr0  = (const float*)d_in[9];
  const float* Wih1  = (const float*)d_in[10];
  const float* Whh1  = (const float*)d_in[11];
  const float* bih1  = (const float*)d_in[12];
  const float* bhh1  = (const float*)d_in[13];
  const float* Whr1  = (const float*)d_in[14];
  float* out = (float*)d_out;

  decoder_scan<<<dim3(BATCH / 16), dim3(256), 0, stream>>>(
      enc, mean_, s, h0, c0,
      Wih0, Whh0, bih0, bhh0, Whr0,
      Wih1, Whh1, bih1, bhh1, Whr1, out);
}